// StackedSelfAttn_FFNN_84619445666614
// MI455X (gfx1250) — compile-verified
//
#include <hip/hip_runtime.h>
#include <math.h>

// ---------------- constants ----------------
namespace {
constexpr int S_    = 8192;
constexpr int E_    = 8;
constexpr int H_    = 2;
constexpr int DH_   = 4;
constexpr int DI_   = 16;
constexpr int L_    = 4;
constexpr int HID_  = 1024;
constexpr float LN_EPS_ = 1e-5f;
constexpr int CH_     = 64;               // i-chunks for the big f1 GEMV
constexpr int CHROWS_ = (S_ * E_) / CH_;  // 1024 activations per chunk
constexpr int WPB_    = 8;                // waves per attention block
constexpr int KCHUNK_ = 512;              // keys staged per LDS chunk
constexpr int NCH_    = S_ / KCHUNK_;     // 16 chunks
}

typedef __attribute__((ext_vector_type(2))) float v2f;
typedef __attribute__((ext_vector_type(8))) float v8f;

__device__ __forceinline__ float gelu_exact(float z) {
  return 0.5f * z * (1.0f + erff(z * 0.70710678118654752f));
}

// async global->LDS copy of 16 bytes per lane (ASYNCcnt-tracked)
__device__ __forceinline__ void g2l_async_b128(const float* src, float* ldsDst) {
  unsigned loff = (unsigned)(uintptr_t)ldsDst;  // low 32 bits of LDS flat addr = LDS offset
  asm volatile("global_load_async_to_lds_b128 %0, %1, off"
               :: "v"(loff), "v"(src)
               : "memory");
}

// ---------------- embedding: x = emb_w[seq] + emb_b + pos ----------------
__global__ void tk_embed(const int* __restrict__ seq,
                         const float* __restrict__ emb_w,
                         const float* __restrict__ emb_b,
                         const float* __restrict__ pos,
                         float* __restrict__ x) {
  int i = blockIdx.x * blockDim.x + threadIdx.x;   // over S_*E_
  int s = i >> 3, e = i & 7;
  int c = seq[s];
  x[i] = emb_w[c * E_ + e] + emb_b[e] + pos[i];
}

// ---------------- QKV projection (E=8, tiny GEMM -> per-element) ----------------
__global__ void tk_qkv(const float* __restrict__ x,
                       const float* __restrict__ wq, const float* __restrict__ bq,
                       const float* __restrict__ wk, const float* __restrict__ bk,
                       const float* __restrict__ wv, const float* __restrict__ bv,
                       float* __restrict__ q, float* __restrict__ k, float* __restrict__ v) {
  int i = blockIdx.x * blockDim.x + threadIdx.x;   // over S_*E_
  int s = i >> 3, e = i & 7;
  const float* xr = x + (size_t)s * E_;
  float aq = bq[e], ak = bk[e], av = bv[e];
#pragma unroll
  for (int c = 0; c < E_; ++c) {
    float xv = xr[c];
    aq += xv * wq[c * E_ + e];
    ak += xv * wk[c * E_ + e];
    av += xv * wv[c * E_ + e];
  }
  q[i] = aq; k[i] = ak; v[i] = av;
}

// ---------------- flash attention -------------------------------------------
// 8 waves per block share one head; K/V chunks double-buffered in LDS via
// global_load_async_to_lds_b128. Per 16-key tile:
//   1 WMMA  : S'[16k x 16q] = K_tile(A,16x4) x Q^T(B,4x16)   (f32, K=DH=4)
//   4 WMMAs : O^T[d x 16q] += V^T chunk (A) x P chunk (B)    (K=4 each)
// Score->B relayout needs only 8 shfl_xor(16) + 8 selects. The next tile's
// LDS operands are prefetched at the top of each iteration so ds latency
// hides under the softmax VALU work.
__global__ void __launch_bounds__(256)
tk_attn(const float* __restrict__ qg,
        const float* __restrict__ kg,
        const float* __restrict__ vg,
        float* __restrict__ og) {
  __shared__ float sK[2][KCHUNK_ * DH_];
  __shared__ float sV[2][KCHUNK_ * DH_];

  const int tid  = threadIdx.x;
  const int wave = tid >> 5;
  const int lane = tid & 31;
  const int n    = lane & 15;      // query column / key row inside tile
  const int h2   = lane >> 4;      // lane half
  const int d4   = lane & 3;       // dh row this lane supplies for V^T A-operand
  const int head = blockIdx.y;
  const int q0   = blockIdx.x * (16 * WPB_) + wave * 16;

  const float scale = 0.5f;        // 1/sqrt(DH)

  // B operand (loop-invariant): B[v+2*h2][n] = Q[q0+n][2*h2+v] * scale
  v2f bq;
  {
    const float* p = qg + (size_t)(q0 + n) * E_ + head * DH_ + 2 * h2;
    bq[0] = p[0] * scale;
    bq[1] = p[1] * scale;
  }

  float m = -INFINITY, lsum = 0.0f;
  v8f oacc = {};   // O^T accumulator: lane-half0 VGPR d (0..3), col n holds O[q0+n][d]

  const float* kbase = kg + head * DH_;
  const float* vbase = vg + head * DH_;

  // stage chunk 0 into buffer 0 (4 async B128 per lane: 2 K rows + 2 V rows)
#pragma unroll
  for (int rep = 0; rep < 2; ++rep) {
    int row = tid + rep * 256;
    g2l_async_b128(kbase + (size_t)row * E_, &sK[0][row * DH_]);
    g2l_async_b128(vbase + (size_t)row * E_, &sV[0][row * DH_]);
  }

  for (int ch = 0; ch < NCH_; ++ch) {
    const int cur = ch & 1;
    if (ch + 1 < NCH_) {
      const float* kn = kbase + (size_t)(ch + 1) * KCHUNK_ * E_;
      const float* vn = vbase + (size_t)(ch + 1) * KCHUNK_ * E_;
#pragma unroll
      for (int rep = 0; rep < 2; ++rep) {
        int row = tid + rep * 256;
        g2l_async_b128(kn + (size_t)row * E_, &sK[1 - cur][row * DH_]);
        g2l_async_b128(vn + (size_t)row * E_, &sV[1 - cur][row * DH_]);
      }
      // 4 newest (next chunk) may stay in flight; current chunk's 4 must be done
      asm volatile("s_wait_asynccnt 0x4" ::: "memory");
    } else {
      asm volatile("s_wait_asynccnt 0x0" ::: "memory");
    }
    __syncthreads();

    const float* Kc = sK[cur];
    const float* Vc = sV[cur];

    // preload tile 0 operands (K A-operand + the 8 V scalars for the 4 PV chunks)
    v2f a_cur;
    {
      const float* kp = Kc + n * DH_ + 2 * h2;
      a_cur[0] = kp[0]; a_cur[1] = kp[1];
    }
    float vA[8];
#pragma unroll
    for (int cj = 0; cj < 4; ++cj) {
      vA[2 * cj + 0] = Vc[(4 * cj + 0 + 2 * h2) * DH_ + d4];
      vA[2 * cj + 1] = Vc[(4 * cj + 1 + 2 * h2) * DH_ + d4];
    }

    for (int kl = 0; kl < KCHUNK_; kl += 16) {
      v8f c = {};
      c = __builtin_amdgcn_wmma_f32_16x16x4_f32(false, a_cur, false, bq,
                                                (short)0, c, false, false);
      // lane view: c[i] = S'[key = i + 8*h2][query = n]

      // prefetch next tile's LDS operands (issued early, consumed next iter)
      v2f a_nxt = a_cur;
      float vN[8];
#pragma unroll
      for (int i = 0; i < 8; ++i) vN[i] = vA[i];
      if (kl + 16 < KCHUNK_) {
        const float* kp = Kc + (kl + 16 + n) * DH_ + 2 * h2;
        a_nxt[0] = kp[0]; a_nxt[1] = kp[1];
#pragma unroll
        for (int cj = 0; cj < 4; ++cj) {
          vN[2 * cj + 0] = Vc[(kl + 16 + 4 * cj + 0 + 2 * h2) * DH_ + d4];
          vN[2 * cj + 1] = Vc[(kl + 16 + 4 * cj + 1 + 2 * h2) * DH_ + d4];
        }
      }

      // --- online softmax (reduction over keys) ---
      float tm = c[0];
#pragma unroll
      for (int i = 1; i < 8; ++i) tm = fmaxf(tm, c[i]);
      tm = fmaxf(tm, __shfl_xor(tm, 16, 32));

      float mn   = fmaxf(m, tm);
      float corr = __expf(m - mn);        // fast native exp: arg <= 0, flush-to-0 OK
      m = mn;
      lsum *= corr;
#pragma unroll
      for (int i = 0; i < 8; ++i) oacc[i] *= corr;

      float p[8], t[8];
      float ps = 0.0f, ts = 0.0f;
#pragma unroll
      for (int i = 0; i < 8; ++i) { p[i] = __expf(c[i] - mn); ps += p[i]; }
#pragma unroll
      for (int i = 0; i < 8; ++i) { t[i] = __shfl_xor(p[i], 16, 32); ts += t[i]; }
      lsum += ps + ts;

      // --- P @ V via 4 WMMAs (K=4 chunks of the 16-key tile) ---
      // B[v+2*h2][n] must equal P[4c+v+2*h2][n]; p[] holds own-half rows,
      // t[] the opposite half's rows.
      v2f va, vb;

      // chunk 0: rows 0..3
      va[0] = vA[0]; va[1] = vA[1];
      vb[0] = h2 ? t[2] : p[0];
      vb[1] = h2 ? t[3] : p[1];
      oacc = __builtin_amdgcn_wmma_f32_16x16x4_f32(false, va, false, vb,
                                                   (short)0, oacc, false, false);
      // chunk 1: rows 4..7
      va[0] = vA[2]; va[1] = vA[3];
      vb[0] = h2 ? t[6] : p[4];
      vb[1] = h2 ? t[7] : p[5];
      oacc = __builtin_amdgcn_wmma_f32_16x16x4_f32(false, va, false, vb,
                                                   (short)0, oacc, false, false);
      // chunk 2: rows 8..11
      va[0] = vA[4]; va[1] = vA[5];
      vb[0] = h2 ? p[2] : t[0];
      vb[1] = h2 ? p[3] : t[1];
      oacc = __builtin_amdgcn_wmma_f32_16x16x4_f32(false, va, false, vb,
                                                   (short)0, oacc, false, false);
      // chunk 3: rows 12..15
      va[0] = vA[6]; va[1] = vA[7];
      vb[0] = h2 ? p[6] : t[4];
      vb[1] = h2 ? p[7] : t[5];
      oacc = __builtin_amdgcn_wmma_f32_16x16x4_f32(false, va, false, vb,
                                                   (short)0, oacc, false, false);

      a_cur = a_nxt;
#pragma unroll
      for (int i = 0; i < 8; ++i) vA[i] = vN[i];
    }
    __syncthreads();   // protect buffer `cur` before it is restaged
  }

  // valid output lives in lane-half 0, accumulator VGPRs 0..3 (O[q=n][d=i])
  if (lane < 16) {
    float inv = 1.0f / lsum;
    float4 r = make_float4(oacc[0] * inv, oacc[1] * inv, oacc[2] * inv, oacc[3] * inv);
    *(float4*)(og + (size_t)(q0 + n) * E_ + head * DH_) = r;
  }
}

// ---------------- output projection + residual + LayerNorm (row-local) --------
__global__ void tk_oproj_ln(const float* xin, const float* __restrict__ ao,
                            const float* __restrict__ wo, const float* __restrict__ bo,
                            float* xout) {
  int s = blockIdx.x * blockDim.x + threadIdx.x;
  if (s >= S_) return;
  float xr[E_], orow[E_], y[E_];
#pragma unroll
  for (int e = 0; e < E_; ++e) { xr[e] = xin[(size_t)s * E_ + e]; orow[e] = ao[(size_t)s * E_ + e]; }
  float mu = 0.0f;
#pragma unroll
  for (int e = 0; e < E_; ++e) {
    float acc = bo[e];
#pragma unroll
    for (int c = 0; c < E_; ++c) acc += orow[c] * wo[c * E_ + e];
    y[e] = xr[e] + acc;
    mu += y[e];
  }
  mu *= (1.0f / E_);
  float var = 0.0f;
#pragma unroll
  for (int e = 0; e < E_; ++e) { float d = y[e] - mu; var += d * d; }
  var *= (1.0f / E_);
  float r = rsqrtf(var + LN_EPS_);
#pragma unroll
  for (int e = 0; e < E_; ++e) xout[(size_t)s * E_ + e] = (y[e] - mu) * r;
}

// ---------------- position-wise FFN + residual + LayerNorm (row-local) --------
__global__ void tk_ffn_ln(const float* xin,
                          const float* __restrict__ w1, const float* __restrict__ b1,
                          const float* __restrict__ w2, const float* __restrict__ b2,
                          float* xout) {
  int s = blockIdx.x * blockDim.x + threadIdx.x;
  if (s >= S_) return;
  float xr[E_], h[DI_], y[E_];
#pragma unroll
  for (int e = 0; e < E_; ++e) xr[e] = xin[(size_t)s * E_ + e];
#pragma unroll
  for (int j = 0; j < DI_; ++j) {
    float acc = b1[j];
#pragma unroll
    for (int e = 0; e < E_; ++e) acc += xr[e] * w1[e * DI_ + j];
    h[j] = gelu_exact(acc);
  }
  float mu = 0.0f;
#pragma unroll
  for (int e = 0; e < E_; ++e) {
    float acc = b2[e];
#pragma unroll
    for (int j = 0; j < DI_; ++j) acc += h[j] * w2[j * E_ + e];
    y[e] = xr[e] + acc;
    mu += y[e];
  }
  mu *= (1.0f / E_);
  float var = 0.0f;
#pragma unroll
  for (int e = 0; e < E_; ++e) { float d = y[e] - mu; var += d * d; }
  var *= (1.0f / E_);
  float r = rsqrtf(var + LN_EPS_);
#pragma unroll
  for (int e = 0; e < E_; ++e) xout[(size_t)s * E_ + e] = (y[e] - mu) * r;
}

// ---------------- big GEMV: partial[c][j] = sum_{i in chunk c} flat[i]*W[i][j] --
// The 256 MB f1_w read is the HBM roofline for the whole problem (~11 us @23.3TB/s).
__global__ void tk_f1_partial(const float* __restrict__ flat,
                              const float* __restrict__ w,
                              float* __restrict__ part) {
  __shared__ float xs[CHROWS_];
  const int tid = threadIdx.x;
  const int j   = blockIdx.x * 256 + tid;       // output column
  const int i0  = blockIdx.y * CHROWS_;         // chunk base row
  for (int t = tid; t < CHROWS_; t += 256) xs[t] = flat[i0 + t];
  __syncthreads();
  float acc = 0.0f;
  const float* wp = w + (size_t)i0 * HID_ + j;
  for (int i = 0; i < CHROWS_; ++i) acc += xs[i] * wp[(size_t)i * HID_];
  part[(size_t)blockIdx.y * HID_ + j] = acc;
}

__global__ void tk_f1_reduce(const float* __restrict__ part,
                             const float* __restrict__ b,
                             float* __restrict__ hid) {
  int j = blockIdx.x * blockDim.x + threadIdx.x;
  float acc = b[j];
  for (int c = 0; c < CH_; ++c) acc += part[(size_t)c * HID_ + j];
  hid[j] = gelu_exact(acc);
}

// ---------------- final tiny GEMV: out[2] = hid @ f2_w + f2_b ----------------
__global__ void tk_f2(const float* __restrict__ hid,
                      const float* __restrict__ w,
                      const float* __restrict__ b,
                      float* __restrict__ out) {
  __shared__ float r0[256], r1[256];
  const int tid = threadIdx.x;
  float a0 = 0.0f, a1 = 0.0f;
  for (int j = tid; j < HID_; j += 256) {
    float hv = hid[j];
    a0 += hv * w[j * 2 + 0];
    a1 += hv * w[j * 2 + 1];
  }
  r0[tid] = a0; r1[tid] = a1;
  __syncthreads();
  for (int off = 128; off > 0; off >>= 1) {
    if (tid < off) { r0[tid] += r0[tid + off]; r1[tid] += r1[tid + off]; }
    __syncthreads();
  }
  if (tid == 0) { out[0] = r0[0] + b[0]; out[1] = r1[0] + b[1]; }
}

// ---------------- launcher ----------------
extern "C" void kernel_launch(void* const* d_in, const int* in_sizes, int n_in,
                              void* d_out, int out_size, void* d_ws, size_t ws_size,
                              hipStream_t stream) {
  const int*   seq   = (const int*)d_in[0];
  const float* emb_w = (const float*)d_in[1];
  const float* emb_b = (const float*)d_in[2];
  const float* pos   = (const float*)d_in[3];
  const float* wq    = (const float*)d_in[4];
  const float* bq    = (const float*)d_in[5];
  const float* wk    = (const float*)d_in[6];
  const float* bk    = (const float*)d_in[7];
  const float* wv    = (const float*)d_in[8];
  const float* bv    = (const float*)d_in[9];
  const float* wo    = (const float*)d_in[10];
  const float* bo    = (const float*)d_in[11];
  const float* w1    = (const float*)d_in[12];
  const float* b1    = (const float*)d_in[13];
  const float* w2    = (const float*)d_in[14];
  const float* b2    = (const float*)d_in[15];
  const float* f1w   = (const float*)d_in[16];
  const float* f1b   = (const float*)d_in[17];
  const float* f2w   = (const float*)d_in[18];
  const float* f2b   = (const float*)d_in[19];
  float* out = (float*)d_out;

  float* ws   = (float*)d_ws;
  float* x    = ws;                  // S*E
  float* qb   = x    + S_ * E_;      // S*E
  float* kb   = qb   + S_ * E_;      // S*E
  float* vb   = kb   + S_ * E_;      // S*E
  float* ao   = vb   + S_ * E_;      // S*E
  float* part = ao   + S_ * E_;      // CH_*HID_
  float* hid  = part + CH_ * HID_;   // HID_

  tk_embed<<<(S_ * E_) / 256, 256, 0, stream>>>(seq, emb_w, emb_b, pos, x);

  for (int l = 0; l < L_; ++l) {
    tk_qkv<<<(S_ * E_) / 256, 256, 0, stream>>>(
        x, wq + l * E_ * E_, bq + l * E_,
           wk + l * E_ * E_, bk + l * E_,
           wv + l * E_ * E_, bv + l * E_,
        qb, kb, vb);
    dim3 ag(S_ / (16 * WPB_), H_);
    tk_attn<<<ag, 256, 0, stream>>>(qb, kb, vb, ao);
    tk_oproj_ln<<<S_ / 256, 256, 0, stream>>>(x, ao, wo + l * E_ * E_, bo + l * E_, x);
    tk_ffn_ln<<<S_ / 256, 256, 0, stream>>>(x, w1 + l * E_ * DI_, b1 + l * DI_,
                                            w2 + l * DI_ * E_, b2 + l * E_, x);
  }

  dim3 g1(HID_ / 256, CH_);
  tk_f1_partial<<<g1, 256, 0, stream>>>(x, f1w, part);
  tk_f1_reduce<<<HID_ / 256, 256, 0, stream>>>(part, f1b, hid);
  tk_f2<<<1, 256, 0, stream>>>(hid, f2w, f2b, out);
}